// StupidIndexator_80556406604429
// MI455X (gfx1250) — compile-verified
//
#include <hip/hip_runtime.h>
#include <math.h>

typedef __attribute__((ext_vector_type(2))) float v2f;
typedef __attribute__((ext_vector_type(8))) float v8f;

#define Q_EXP 60
#define M_HKL 80
#define NROT 64000
#define WAVES_PER_BLK 8
#define NB 125            // scatter/hist blocks: 125 * 512 = 64000
#define BS 512
#define NBINS 61          // rate in 0..60
#define COS_THR 0.9992001066609779f   // cos(0.04)

// ---------------------------------------------------------------------------
// Kernel 1: one wave32 per rotation. Build R, rotate basis, make 80 theo unit
// rays (staged in LDS), then dots(60x80, K=3) via V_WMMA_F32_16X16X4_F32.
// Matching uses ballot: row matched <=> any lane's |dot|-max >= cos_thr.
// ---------------------------------------------------------------------------
__global__ __launch_bounds__(256) void rate_kernel(
    const float* __restrict__ u_q, const float* __restrict__ recip,
    const float* __restrict__ hkl, const float* __restrict__ angles,
    int* __restrict__ rateI)
{
    __shared__ float ldsU[WAVES_PER_BLK][M_HKL * 3];
    const int w    = threadIdx.x >> 5;
    const int lane = threadIdx.x & 31;
    const int g    = blockIdx.x * WAVES_PER_BLK + w;   // rotation index

    // ---- rotation matrix R = Rz(t3) Ry(t2) Rx(t1) (uniform across wave)
    float t1 = angles[g * 3 + 0], t2 = angles[g * 3 + 1], t3 = angles[g * 3 + 2];
    float s1, c1, s2, c2, s3, c3;
    sincosf(t1, &s1, &c1);
    sincosf(t2, &s2, &c2);
    sincosf(t3, &s3, &c3);
    float R00 = c3 * c2, R01 = c3 * s2 * s1 - s3 * c1, R02 = c3 * s2 * c1 + s3 * s1;
    float R10 = s3 * c2, R11 = s3 * s2 * s1 + c3 * c1, R12 = s3 * s2 * c1 - c3 * s1;
    float R20 = -s2,     R21 = c2 * s1,                R22 = c2 * c1;

    // ---- M = R @ reciprocal (row-major 3x3)
    float r0 = recip[0], r1 = recip[1], r2 = recip[2];
    float r3 = recip[3], r4 = recip[4], r5 = recip[5];
    float r6 = recip[6], r7 = recip[7], r8 = recip[8];
    float M00 = R00 * r0 + R01 * r3 + R02 * r6;
    float M01 = R00 * r1 + R01 * r4 + R02 * r7;
    float M02 = R00 * r2 + R01 * r5 + R02 * r8;
    float M10 = R10 * r0 + R11 * r3 + R12 * r6;
    float M11 = R10 * r1 + R11 * r4 + R12 * r7;
    float M12 = R10 * r2 + R11 * r5 + R12 * r8;
    float M20 = R20 * r0 + R21 * r3 + R22 * r6;
    float M21 = R20 * r1 + R21 * r4 + R22 * r7;
    float M22 = R20 * r2 + R21 * r5 + R22 * r8;

    // ---- theoretical unit rays -> per-wave LDS scratch
    for (int m = lane; m < M_HKL; m += 32) {
        float hx = hkl[m * 3 + 0], hy = hkl[m * 3 + 1], hz = hkl[m * 3 + 2];
        float qx = M00 * hx + M01 * hy + M02 * hz;
        float qy = M10 * hx + M11 * hy + M12 * hz;
        float qz = M20 * hx + M21 * hy + M22 * hz;
        float inv = rsqrtf(qx * qx + qy * qy + qz * qz);
        ldsU[w][m * 3 + 0] = qx * inv;
        ldsU[w][m * 3 + 1] = qy * inv;
        ldsU[w][m * 3 + 2] = qz * inv;
    }
    __syncthreads();

    const int  sub  = lane & 15;
    const bool hi   = lane >= 16;
    const int  koff = hi ? 2 : 0;        // K-component offset for fragment reg0

    // ---- A fragments: u_q rows (16x4 f32 per tile; K=3 padded with 0)
    // layout: reg0 = K0 (lo half) / K2 (hi half); reg1 = K1 / K3
    // Branchless: clamp row index (always in-bounds), select-zero padded rows.
    v2f Afrag[4];
#pragma unroll
    for (int t = 0; t < 4; ++t) {
        int q  = t * 16 + sub;
        int qc = (q < Q_EXP) ? q : (Q_EXP - 1);
        float l0 = u_q[qc * 3 + koff];
        float l1 = u_q[qc * 3 + 1];
        bool  ok = (q < Q_EXP);
        Afrag[t][0] = ok ? l0 : 0.f;
        Afrag[t][1] = (ok && !hi) ? l1 : 0.f;
    }

    // ---- B fragments: theo ray columns (4x16 f32 per tile), branchless
    v2f Bfrag[5];
#pragma unroll
    for (int j = 0; j < 5; ++j) {
        int m = j * 16 + sub;
        float l0 = ldsU[w][m * 3 + koff];
        float l1 = ldsU[w][m * 3 + 1];
        Bfrag[j][0] = l0;
        Bfrag[j][1] = hi ? 0.f : l1;
    }

    // ---- 20 WMMAs: D[q][m] = u_q[q] . U[m]; fold abs+max over m-tiles, then
    // count matched rows with ballot (no cross-lane data movement needed).
    int rate = 0;
#pragma unroll
    for (int t = 0; t < 4; ++t) {
        v8f mx = {0.f, 0.f, 0.f, 0.f, 0.f, 0.f, 0.f, 0.f};
#pragma unroll
        for (int j = 0; j < 5; ++j) {
            v8f c = {0.f, 0.f, 0.f, 0.f, 0.f, 0.f, 0.f, 0.f};
            v8f d = __builtin_amdgcn_wmma_f32_16x16x4_f32(
                false, Afrag[t], false, Bfrag[j], (short)0, c, false, false);
#pragma unroll
            for (int r = 0; r < 8; ++r)
                mx[r] = fmaxf(mx[r], fabsf(d[r]));
        }
        // reg r: lanes 0-15 hold row t*16+r (m = tile*16 + lane),
        //        lanes 16-31 hold row t*16+r+8.
        // Row matched <=> some lane in its half has mx >= thr.
#pragma unroll
        for (int r = 0; r < 8; ++r) {
            unsigned b = __builtin_amdgcn_ballot_w32(mx[r] >= COS_THR);
            rate += ((b & 0xFFFFu) != 0u) ? 1 : 0;   // row t*16+r
            rate += ((b >> 16)    != 0u) ? 1 : 0;    // row t*16+r+8
        }
    }
    if (lane == 0) rateI[g] = rate;   // zero-padded rows (q>=60) never match
}

// ---------------------------------------------------------------------------
// Kernel 2: per-block histograms of rate values (bins 0..60)
// ---------------------------------------------------------------------------
__global__ __launch_bounds__(BS) void hist_kernel(const int* __restrict__ rateI,
                                                  int* __restrict__ hist)
{
    __shared__ int h[NBINS];
    for (int i = threadIdx.x; i < NBINS; i += BS) h[i] = 0;
    __syncthreads();
    int g = blockIdx.x * BS + threadIdx.x;
    int v = rateI[g];
    atomicAdd(&h[v], 1);
    __syncthreads();
    for (int i = threadIdx.x; i < NBINS; i += BS)
        hist[i * NB + blockIdx.x] = h[i];
}

// ---------------------------------------------------------------------------
// Kernel 3: single-block exclusive scan over (bin descending, block ascending)
// -> stable descending counting-sort offsets
// ---------------------------------------------------------------------------
__global__ __launch_bounds__(1024) void scan_kernel(const int* __restrict__ hist,
                                                    int* __restrict__ offs)
{
    const int N  = NBINS * NB;           // 7625
    const int CE = (N + 1023) / 1024;    // 8
    __shared__ int partial[1024];
    int base = threadIdx.x * CE;
    int vals[8];
    int s = 0;
    for (int k = 0; k < CE; ++k) {
        int p = base + k;
        int c = 0;
        if (p < N) {
            int bin = 60 - p / NB;
            int blk = p - (p / NB) * NB;
            c = hist[bin * NB + blk];
        }
        vals[k] = c;
        s += c;
    }
    partial[threadIdx.x] = s;
    __syncthreads();
    for (int off = 1; off < 1024; off <<= 1) {
        int x = (threadIdx.x >= off) ? partial[threadIdx.x - off] : 0;
        __syncthreads();
        partial[threadIdx.x] += x;
        __syncthreads();
    }
    int run = (threadIdx.x > 0) ? partial[threadIdx.x - 1] : 0;
    for (int k = 0; k < CE; ++k) {
        int p = base + k;
        if (p < N) {
            int bin = 60 - p / NB;
            int blk = p - (p / NB) * NB;
            offs[bin * NB + blk] = run;
        }
        run += vals[k];
    }
}

// ---------------------------------------------------------------------------
// Kernel 4: stable scatter (intra-block rank over original order)
// ---------------------------------------------------------------------------
__global__ __launch_bounds__(BS) void scatter_kernel(
    const int* __restrict__ rateI, const int* __restrict__ offs,
    const float* __restrict__ angles,
    float* __restrict__ outAng, float* __restrict__ outRate)
{
    __shared__ int sh[BS];
    int g = blockIdx.x * BS + threadIdx.x;
    int v = rateI[g];
    sh[threadIdx.x] = v;
    __syncthreads();
    int rank = 0;
    for (int j = 0; j < threadIdx.x; ++j)
        rank += (sh[j] == v) ? 1 : 0;
    int pos = offs[v * NB + blockIdx.x] + rank;
    outAng[pos * 3 + 0] = angles[g * 3 + 0];
    outAng[pos * 3 + 1] = angles[g * 3 + 1];
    outAng[pos * 3 + 2] = angles[g * 3 + 2];
    outRate[pos] = (float)v;
}

// ---------------------------------------------------------------------------
extern "C" void kernel_launch(void* const* d_in, const int* in_sizes, int n_in,
                              void* d_out, int out_size, void* d_ws, size_t ws_size,
                              hipStream_t stream)
{
    const float* u_q    = (const float*)d_in[0];   // (60,3)
    const float* recip  = (const float*)d_in[1];   // (3,3)
    const float* hkl    = (const float*)d_in[2];   // (80,3)
    const float* angles = (const float*)d_in[3];   // (64000,3)

    float* outAng  = (float*)d_out;                // 64000*3 sorted angles
    float* outRate = outAng + NROT * 3;            // 64000 sorted rates

    int* rateI = (int*)d_ws;                       // 64000 ints
    int* hist  = rateI + NROT;                     // 61*125 ints
    int* offs  = hist + NBINS * NB;                // 61*125 ints

    rate_kernel<<<NROT / WAVES_PER_BLK, 256, 0, stream>>>(u_q, recip, hkl, angles, rateI);
    hist_kernel<<<NB, BS, 0, stream>>>(rateI, hist);
    scan_kernel<<<1, 1024, 0, stream>>>(hist, offs);
    scatter_kernel<<<NB, BS, 0, stream>>>(rateI, offs, angles, outAng, outRate);
}